// Encoder_17386027614470
// MI455X (gfx1250) — compile-verified
//
#include <hip/hip_runtime.h>

typedef float v2f __attribute__((ext_vector_type(2)));
typedef float v8f __attribute__((ext_vector_type(8)));

#define GDIM 64

__global__ void zero_f32(float* __restrict__ p, long long n) {
  long long i = (long long)blockIdx.x * blockDim.x + threadIdx.x;
  if (i < n) p[i] = 0.f;
}

__global__ void copy_f32(float* __restrict__ d, const float* __restrict__ s, long long n) {
  long long i = (long long)blockIdx.x * blockDim.x + threadIdx.x;
  if (i < n) d[i] = s[i];
}

// agg[dst] += h[src], float4 per thread, shift = log2(F/4)
__global__ void edge_scatter(const float* __restrict__ h, const int* __restrict__ src,
                             const int* __restrict__ dst, float* __restrict__ agg,
                             int E, int shift, int F) {
  long long t = (long long)blockIdx.x * blockDim.x + threadIdx.x;
  long long total = (long long)E << shift;
  if (t >= total) return;
  int e = (int)(t >> shift);
  int c = (int)(t & ((1 << shift) - 1));
  int s = src[e], d = dst[e];
  const float4 v = *reinterpret_cast<const float4*>(h + (size_t)s * F + c * 4);
  float* p = agg + (size_t)d * F + c * 4;
  atomicAdd(p + 0, v.x);
  atomicAdd(p + 1, v.y);
  atomicAdd(p + 2, v.z);
  atomicAdd(p + 3, v.w);
}

// out(N x 64) = ReLU(A(N x din) @ W(din x 64) + bias), fp32 WMMA 16x16x4.
// Block = 128 threads (4 waves); wave w owns columns [16w, 16w+16); block owns 16 rows.
__global__ __launch_bounds__(128) void gemm_bias_relu_wmma(
    const float* __restrict__ A, const float* __restrict__ W,
    const float* __restrict__ bias, float* __restrict__ out, int N, int din) {
  __shared__ float wsh[128 * GDIM];  // 32 KB max (din<=128)
  const int tid = threadIdx.x;
  for (int i = tid; i < din * GDIM; i += 128) wsh[i] = W[i];
  __syncthreads();

  const int lane = tid & 31;
  const int wave = tid >> 5;
  const int colbase = wave * 16;
  const int n = lane & 15;        // N index for B/C/D, M index for A
  const int half = lane >> 4;     // selects K pair (A/B) / row group (C/D)
  const int rows = blockIdx.x * 16;
  int arow = rows + n;
  if (arow >= N) arow = N - 1;    // clamp (N is a multiple of 16 here anyway)
  const float* arow_ptr = A + (size_t)arow * din + half * 2;

  const float bval = bias[colbase + n];
  v8f acc = {bval, bval, bval, bval, bval, bval, bval, bval};

  for (int k0 = 0; k0 < din; k0 += 4) {
    // A 16x4 frag: lane holds A[m][half*2 + {0,1}] (8B-aligned since din%4==0)
    v2f a = *reinterpret_cast<const v2f*>(arow_ptr + k0);
    // B 4x16 frag: lane holds W[k0+half*2 + {0,1}][colbase+n]
    const int krow = k0 + half * 2;
    v2f b;
    b.x = wsh[krow * GDIM + colbase + n];
    b.y = wsh[(krow + 1) * GDIM + colbase + n];
    acc = __builtin_amdgcn_wmma_f32_16x16x4_f32(
        /*neg_a=*/false, a, /*neg_b=*/false, b,
        /*c_mod=*/(short)0, acc, /*reuse_a=*/false, /*reuse_b=*/false);
  }

#pragma unroll
  for (int r = 0; r < 8; ++r) {
    int row = rows + r + half * 8;  // C/D layout: VGPR r -> M = r + 8*half
    if (row < N) {
      float v = acc[r];
      out[(size_t)row * GDIM + colbase + n] = v > 0.f ? v : 0.f;
    }
  }
}

// Column sums / sums-of-squares of h (N x 64) into stats[0..63]=sum, [64..127]=sumsq
__global__ void bn_stats(const float* __restrict__ h, float* __restrict__ stats, int N) {
  __shared__ float s1[256];
  __shared__ float s2[256];
  const int tid = threadIdx.x;
  const int col = tid & 63;
  const int rl = tid >> 6;  // 0..3
  float s = 0.f, ss = 0.f;
  for (int row = blockIdx.x * 4 + rl; row < N; row += gridDim.x * 4) {
    float v = h[(size_t)row * GDIM + col];
    s += v;
    ss += v * v;
  }
  s1[tid] = s;
  s2[tid] = ss;
  __syncthreads();
  if (rl == 0) {
    s = s1[col] + s1[64 + col] + s1[128 + col] + s1[192 + col];
    ss = s2[col] + s2[64 + col] + s2[128 + col] + s2[192 + col];
    atomicAdd(&stats[col], s);
    atomicAdd(&stats[64 + col], ss);
  }
}

__global__ void bn_finalize(const float* __restrict__ stats, const float* __restrict__ gamma,
                            const float* __restrict__ beta, float* __restrict__ scsh, int N) {
  const int c = threadIdx.x;  // 64 threads
  const float inv_n = 1.0f / (float)N;
  const float mean = stats[c] * inv_n;
  const float var = stats[64 + c] * inv_n - mean * mean;  // biased variance
  const float sc = gamma[c] * rsqrtf(var + 1e-5f);
  scsh[c] = sc;
  scsh[64 + c] = beta[c] - mean * sc;
}

// h_out = BN(h_in); also writes xs column-slice and atomically add-pools per graph.
__global__ void bn_apply_pool(const float* __restrict__ hin, const float* __restrict__ scsh,
                              const int* __restrict__ batch, float* __restrict__ hout,
                              float* __restrict__ xs, float* __restrict__ pool,
                              int N, int layer_off) {
  long long i = (long long)blockIdx.x * blockDim.x + threadIdx.x;
  if (i >= (long long)N * GDIM) return;
  const int row = (int)(i >> 6);
  const int col = (int)(i & 63);
  const float v = hin[i] * scsh[col] + scsh[64 + col];
  hout[i] = v;
  xs[(size_t)row * 192 + layer_off + col] = v;
  atomicAdd(&pool[(size_t)batch[row] * 192 + layer_off + col], v);
}

extern "C" void kernel_launch(void* const* d_in, const int* in_sizes, int n_in,
                              void* d_out, int out_size, void* d_ws, size_t ws_size,
                              hipStream_t stream) {
  const float* x = (const float*)d_in[0];
  const int* edges = (const int*)d_in[1];   // (2, E): src then dst
  const int* batch = (const int*)d_in[2];
  const int N = in_sizes[0] / 128;
  const int E = in_sizes[1] / 2;
  const int G = (out_size - N * 192) / 192;
  const int* src = edges;
  const int* dst = edges + E;

  float* ws = (float*)d_ws;
  float* agg = ws;                          // N*128
  float* t1 = agg + (size_t)N * 128;        // N*64
  float* t2 = t1 + (size_t)N * 64;          // N*64
  float* hcur = t2 + (size_t)N * 64;        // N*64
  float* stats = hcur + (size_t)N * 64;     // 128
  float* scsh = stats + 128;                // 128

  float* pool = (float*)d_out;              // G*192
  float* xs = pool + (size_t)G * 192;       // N*192

  const long long pn = (long long)G * 192;
  zero_f32<<<(unsigned)((pn + 255) / 256), 256, 0, stream>>>(pool, pn);

  for (int layer = 0; layer < 3; ++layer) {
    const float* w1 = (const float*)d_in[4 + layer * 6 + 0];
    const float* b1 = (const float*)d_in[4 + layer * 6 + 1];
    const float* w2 = (const float*)d_in[4 + layer * 6 + 2];
    const float* b2 = (const float*)d_in[4 + layer * 6 + 3];
    const float* gm = (const float*)d_in[4 + layer * 6 + 4];
    const float* bt = (const float*)d_in[4 + layer * 6 + 5];
    const int din = (layer == 0) ? 128 : 64;
    const float* hin = (layer == 0) ? x : hcur;

    // agg = h (self term), then agg[dst] += h[src] over all edges
    const long long an = (long long)N * din;
    copy_f32<<<(unsigned)((an + 255) / 256), 256, 0, stream>>>(agg, hin, an);
    const int shift = (din == 128) ? 5 : 4;  // log2(din/4)
    const long long et = (long long)E << shift;
    edge_scatter<<<(unsigned)((et + 255) / 256), 256, 0, stream>>>(hin, src, dst, agg, E, shift, din);

    // MLP: ReLU(agg@w1+b1) -> ReLU(..@w2+b2)
    const int gridM = (N + 15) / 16;
    gemm_bias_relu_wmma<<<gridM, 128, 0, stream>>>(agg, w1, b1, t1, N, din);
    gemm_bias_relu_wmma<<<gridM, 128, 0, stream>>>(t1, w2, b2, t2, N, GDIM);

    // BatchNorm (training-mode batch stats, biased var)
    zero_f32<<<1, 128, 0, stream>>>(stats, 128);
    bn_stats<<<512, 256, 0, stream>>>(t2, stats, N);
    bn_finalize<<<1, 64, 0, stream>>>(stats, gm, bt, scsh, N);

    // Apply BN, emit xs slice, pool per graph
    const long long hn = (long long)N * GDIM;
    bn_apply_pool<<<(unsigned)((hn + 255) / 256), 256, 0, stream>>>(
        t2, scsh, batch, hcur, xs, pool, N, layer * GDIM);
  }
}